// HybridQuantumNetQML_65481071397824
// MI455X (gfx1250) — compile-verified
//
#include <hip/hip_runtime.h>

typedef __attribute__((ext_vector_type(2))) float v2f;
typedef __attribute__((ext_vector_type(8))) float v8f;

#define WAVES          2
#define THREADS        (WAVES * 32)
#define ROWS_PER_BLOCK (WAVES * 16)   // 32 samples per block
#define KDIM           787
#define KPAD           788            // multiple of 4 for the x4 WMMA K-loop
#define ASTRIDE        790            // even (8B-aligned pairs), 790%64=22 -> conflict-free rows
#define NDIM           10

__global__ __launch_bounds__(THREADS)
void hqn_fused_kernel(const float* __restrict__ x,     // (B,1,28,28)
                      const float* __restrict__ W,     // (787,10)
                      const float* __restrict__ bias,  // (10)
                      const float* __restrict__ sw,    // (4)
                      float* __restrict__ out)         // (B,10)
{
    __shared__ float sA[WAVES][16][ASTRIDE];  // combined features per wave-tile
    __shared__ float sWT[16][ASTRIDE];        // transposed zero-padded weights
    __shared__ float sC[WAVES][16][17];       // logits scratch

    const int t       = threadIdx.x;
    const int wave    = t >> 5;
    const int lane    = t & 31;
    const int rowbase = blockIdx.x * ROWS_PER_BLOCK;

    // ---- stage transposed, zero-padded weights: sWT[n][k] = W[k*10+n] ----
    #pragma unroll 1
    for (int i = 0; i < (16 * KPAD) / THREADS; ++i) {   // 197 iters
        int idx = t + i * THREADS;
        int n = idx / KPAD, k = idx % KPAD;
        float v = 0.0f;
        if (n < NDIM && k < KDIM) v = W[k * NDIM + n];
        sWT[n][k] = v;
    }

    // ---- quanvolution features: 32 samples x 196 patches = 6272 tasks ----
    #pragma unroll 1
    for (int i = 0; i < (ROWS_PER_BLOCK * 196) / THREADS; ++i) {  // 98 iters
        int task = t + i * THREADS;
        int s = task / 196, p = task % 196;
        int pi = p / 14, pj = p % 14;
        const float* px = x + (size_t)(rowbase + s) * 784 + (2 * pi) * 28 + 2 * pj;
        float a = px[0], b = px[1], c = px[28], d = px[29];
        float ca = __cosf(a), cb = __cosf(b), cc = __cosf(c), cd = __cosf(d);
        float* dst = &sA[s >> 4][s & 15][4 * p];
        dst[0] = ca; dst[1] = ca * cb; dst[2] = cc; dst[3] = cc * cd;
    }
    __syncthreads();

    // ---- extras per sample: k=784 qfc, k=785/786 sampler softmax, k=787 pad ----
    if (t < ROWS_PER_BLOCK) {
        int wv = t >> 4, r = t & 15;
        float f0 = sA[wv][r][0], f1 = sA[wv][r][1];
        float f2 = sA[wv][r][2], f3 = sA[wv][r][3];
        float c0 = __cosf(f0), c1 = __cosf(f1), c2 = __cosf(f2), c3 = __cosf(f3);
        float qfc = (c0 + c0 * c1 + c2 + c2 * c3) * 0.25f;
        float t0 = f0 + sw[0] + sw[2];
        float t1 = f1 + sw[1] + sw[3];
        float z0 = __cosf(t0);
        float z1 = z0 * __cosf(t1);
        float m  = fmaxf(z0, z1);
        float e0 = __expf(z0 - m), e1 = __expf(z1 - m);
        float inv = 1.0f / (e0 + e1);
        sA[wv][r][784] = qfc;
        sA[wv][r][785] = e0 * inv;
        sA[wv][r][786] = e1 * inv;
        sA[wv][r][787] = 0.0f;
    }
    __syncthreads();

    // ---- fp32 WMMA GEMM: combined(16x788) @ Wpad(788x16), K-steps of 4 ----
    // A 16x4 layout: lanes 0-15 hold M=lane, K={0,1}; lanes 16-31 K={2,3}.
    // B 4x16 mirrored: lane%16 = column, same K split.
    const int arow = lane & 15;
    const int koff = (lane >> 4) * 2;
    const float* aptr = &sA[wave][arow][koff];
    const float* bptr = &sWT[arow][koff];
    v8f acc = {};
    #pragma unroll 4
    for (int kk = 0; kk < KPAD / 4; ++kk) {   // 197 WMMAs per wave
        v2f av = *reinterpret_cast<const v2f*>(aptr + 4 * kk);
        v2f bv = *reinterpret_cast<const v2f*>(bptr + 4 * kk);
        acc = __builtin_amdgcn_wmma_f32_16x16x4_f32(false, av, false, bv,
                                                    (short)0, acc, false, false);
    }

    // ---- spill C tile (lane l holds col l%16 of rows v+8*(l/16)) ----
    #pragma unroll
    for (int v = 0; v < 8; ++v)
        sC[wave][v + 8 * (lane >> 4)][lane & 15] = acc[v];
    __syncthreads();

    // ---- epilogue: bias + log_softmax per row, write (B,10) ----
    if (lane < 16) {
        int grow = rowbase + wave * 16 + lane;
        float logit[NDIM];
        float m = -INFINITY;
        #pragma unroll
        for (int n = 0; n < NDIM; ++n) {
            logit[n] = sC[wave][lane][n] + bias[n];
            m = fmaxf(m, logit[n]);
        }
        float ssum = 0.0f;
        #pragma unroll
        for (int n = 0; n < NDIM; ++n) ssum += __expf(logit[n] - m);
        float lse = m + __logf(ssum);
        #pragma unroll
        for (int n = 0; n < NDIM; ++n)
            out[(size_t)grow * NDIM + n] = logit[n] - lse;
    }
}

extern "C" void kernel_launch(void* const* d_in, const int* in_sizes, int n_in,
                              void* d_out, int out_size, void* d_ws, size_t ws_size,
                              hipStream_t stream) {
    (void)n_in; (void)out_size; (void)d_ws; (void)ws_size;
    const float* x    = (const float*)d_in[0];   // (B,1,28,28)
    const float* W    = (const float*)d_in[1];   // (787,10)
    const float* bias = (const float*)d_in[2];   // (10)
    const float* sw   = (const float*)d_in[3];   // (4)
    float* out = (float*)d_out;

    int B = in_sizes[0] / 784;                   // 32768
    int grid = B / ROWS_PER_BLOCK;               // 1024
    hqn_fused_kernel<<<grid, THREADS, 0, stream>>>(x, W, bias, sw, out);
}